// GraphConvolution_17343077941573
// MI455X (gfx1250) — compile-verified
//
#include <hip/hip_runtime.h>

typedef __attribute__((ext_vector_type(2))) float v2f;
typedef __attribute__((ext_vector_type(8))) float v8f;

#define IN_F  128
#define OUT_F 128

// ---------------------------------------------------------------------------
// Kernel 0: out[n][o] = bias[o]   (out buffer is poisoned by the harness)
// ---------------------------------------------------------------------------
__global__ __launch_bounds__(256) void rgcn_init_bias(
    float* __restrict__ out, const float* __restrict__ bias, int total)
{
    int i = blockIdx.x * blockDim.x + threadIdx.x;
    if (i < total) out[i] = bias[i & (OUT_F - 1)];
}

// ---------------------------------------------------------------------------
// Kernel 1: feats[N,128] = x[N,128] @ W[128,128] via V_WMMA_F32_16X16X4_F32
// One wave per 16-row stripe; 8 column tiles of 16; K-loop of 32 x (8 WMMA).
// Layouts (ISA 7.12.2, f32):
//   A 16x4 : lanes0-15 hold row m0+l16, VGPR0=K0 VGPR1=K1; lanes16-31 K2/K3
//   B 4x16 : lanes0-15 hold col n0+l16, VGPR0=K0 VGPR1=K1; lanes16-31 K2/K3
//   D 16x16: VGPR v, lanes0-15 -> row m0+v, lanes16-31 -> row m0+v+8
// ---------------------------------------------------------------------------
__global__ __launch_bounds__(256) void rgcn_gemm_wmma(
    const float* __restrict__ x, const float* __restrict__ w,
    float* __restrict__ feats, int n)
{
    const int wave = blockIdx.x * (blockDim.x >> 5) + (threadIdx.x >> 5);
    const int lane = threadIdx.x & 31;
    const int half = lane >> 4;      // 0: K{0,1}, 1: K{2,3}
    const int l16  = lane & 15;

    const int m0 = wave * 16;
    if (m0 >= n) return;

    int rowA = m0 + l16;
    if (rowA >= n) rowA = n - 1;     // clamp for the tail tile (EXEC stays full)

    v8f acc[8];
    #pragma unroll
    for (int t = 0; t < 8; ++t) {
        v8f z = {0.f, 0.f, 0.f, 0.f, 0.f, 0.f, 0.f, 0.f};
        acc[t] = z;
    }

    const float* aptr = x + (size_t)rowA * IN_F + 2 * half;

    for (int k = 0; k < IN_F; k += 4) {
        v2f a = *(const v2f*)(aptr + k);   // 8-byte aligned: k%4==0, 2*half even
        const int kb = k + 2 * half;
        #pragma unroll
        for (int t = 0; t < 8; ++t) {
            const int col = t * 16 + l16;
            v2f b;
            b.x = w[(size_t)(kb + 0) * OUT_F + col];
            b.y = w[(size_t)(kb + 1) * OUT_F + col];
            acc[t] = __builtin_amdgcn_wmma_f32_16x16x4_f32(
                /*neg_a=*/false, a, /*neg_b=*/false, b,
                /*c_mod=*/(short)0, acc[t], /*reuse_a=*/false, /*reuse_b=*/false);
        }
    }

    #pragma unroll
    for (int t = 0; t < 8; ++t) {
        #pragma unroll
        for (int v = 0; v < 8; ++v) {
            const int row = m0 + v + 8 * half;
            if (row < n)
                feats[(size_t)row * OUT_F + t * 16 + l16] = acc[t][v];
        }
    }
}

// ---------------------------------------------------------------------------
// Kernel 2: per-edge scatter. One wave per edge, one float4 per lane.
//   alp  = alpha[et[e]] + alpha[et[partner(e)]]
//   out[dst[e]][:] += feats[src[e]][:] * alp   (f32 atomics; out lives in L2)
// partner(e): e<T -> e+T ; T<=e<2T -> e-T ; else e   (self-loops)
// ---------------------------------------------------------------------------
__global__ __launch_bounds__(256) void rgcn_scatter(
    const float* __restrict__ feats, const int* __restrict__ et,
    const int* __restrict__ src, const int* __restrict__ dst,
    const float* __restrict__ alpha, float* __restrict__ out,
    int e, int t)
{
    const int edge = blockIdx.x * (blockDim.x >> 5) + (threadIdx.x >> 5);
    if (edge >= e) return;
    const int lane = threadIdx.x & 31;

    const int partner = (edge < t) ? (edge + t)
                       : (edge < 2 * t) ? (edge - t) : edge;
    const int e1 = et[edge];
    const int e2 = et[partner];
    const float a1 = (e1 == 0) ? 0.f : alpha[e1];   // padding_idx = 0
    const float a2 = (e2 == 0) ? 0.f : alpha[e2];
    const float alp = a1 + a2;

    const int s = src[edge];
    const int d = dst[edge];

    const float4 f = *(const float4*)(feats + (size_t)s * OUT_F + lane * 4);
    float* op = out + (size_t)d * OUT_F + lane * 4;
    atomicAdd(op + 0, f.x * alp);
    atomicAdd(op + 1, f.y * alp);
    atomicAdd(op + 2, f.z * alp);
    atomicAdd(op + 3, f.w * alp);
}

// ---------------------------------------------------------------------------
extern "C" void kernel_launch(void* const* d_in, const int* in_sizes, int n_in,
                              void* d_out, int out_size, void* d_ws, size_t ws_size,
                              hipStream_t stream)
{
    (void)n_in; (void)out_size; (void)ws_size;

    const float* x     = (const float*)d_in[0];
    const int*   et    = (const int*)  d_in[1];
    const int*   src   = (const int*)  d_in[2];
    const int*   dst   = (const int*)  d_in[3];
    const float* w     = (const float*)d_in[4];
    const float* alpha = (const float*)d_in[5];
    const float* bias  = (const float*)d_in[6];

    const int outf = in_sizes[6];             // 128
    const int inf  = in_sizes[4] / outf;      // 128
    const int n    = in_sizes[0] / inf;       // num nodes
    const int e    = in_sizes[1];             // total edges
    const int t    = (e - n) / 2;             // forward edge count

    float* feats = (float*)d_ws;              // N*OUT_F floats (~51.2 MB)
    float* out   = (float*)d_out;

    // out = bias
    {
        const int total   = n * outf;
        const int threads = 256;
        const int blocks  = (total + threads - 1) / threads;
        rgcn_init_bias<<<blocks, threads, 0, stream>>>(out, bias, total);
    }
    // feats = x @ W   (WMMA f32)
    {
        const int rowTiles      = (n + 15) / 16;   // one wave per tile
        const int wavesPerBlock = 8;               // 256 threads
        const int blocks        = (rowTiles + wavesPerBlock - 1) / wavesPerBlock;
        rgcn_gemm_wmma<<<blocks, 256, 0, stream>>>(x, w, feats, n);
    }
    // out[dst] += feats[src] * alp
    {
        const int wavesPerBlock = 8;
        const int blocks        = (e + wavesPerBlock - 1) / wavesPerBlock;
        rgcn_scatter<<<blocks, 256, 0, stream>>>(feats, et, src, dst, alpha,
                                                 out, e, t);
    }
}